// Controller_64364379898395
// MI455X (gfx1250) — compile-verified
//
#include <hip/hip_runtime.h>
#include <hip/hip_bf16.h>

// ---------------------------------------------------------------------------
// ENAS-style LSTM controller for MI455X (gfx1250, wave32).
//
// Per-step recurrent GEMV (Whh[8192x2048] @ h) is the dominant cost.
// op_Whh + sk_Whh = 128 MB -> resident in the 192 MB L2 after first pass.
// fp32 WMMA (V_WMMA_F32_16X16X4_F32) computes the GEMV: A = h replicated
// across the 16 M-rows, B = 16 weight rows; D rows are all identical and
// equal the 16 partial dot products. Four independent accumulators break
// the serial WMMA->WMMA C-chain (dep distance 4 under unroll 8).
// ---------------------------------------------------------------------------

#define H    2048
#define G4   8192          // 4*H gate width
#define NUM_NODES 24
#define DIM_W 512
#define NUM_OPS 5

typedef __attribute__((ext_vector_type(2))) float v2f;
typedef __attribute__((ext_vector_type(8))) float v8f;

// ---------------------------------------------------------------- utilities
__global__ void zero_kernel(float* __restrict__ p, int n) {
  int i = blockIdx.x * blockDim.x + threadIdx.x;
  if (i < n) p[i] = 0.0f;
}

__device__ __forceinline__ float wave_reduce(float s) {
  #pragma unroll
  for (int off = 16; off; off >>= 1) s += __shfl_xor(s, off, 32);
  return s;
}

// ------------------------------------------------ precompute x@Wih.T + biases
// One wave per output row; rows [0,8192) -> op cell, [8192,16384) -> sk cell.
__global__ __launch_bounds__(256) void xb_kernel(
    const float* __restrict__ g_emb,
    const float* __restrict__ op_Wih, const float* __restrict__ op_bih,
    const float* __restrict__ op_bhh,
    const float* __restrict__ sk_Wih, const float* __restrict__ sk_bih,
    const float* __restrict__ sk_bhh,
    float* __restrict__ xb_op, float* __restrict__ xb_sk) {
  int gw   = (blockIdx.x * blockDim.x + threadIdx.x) >> 5;   // global wave id
  int lane = threadIdx.x & 31;
  bool sk  = gw >= G4;
  int row  = sk ? gw - G4 : gw;
  const float4* Wv = (const float4*)((sk ? sk_Wih : op_Wih) + (size_t)row * H);
  const float4* xv = (const float4*)g_emb;
  float s = 0.0f;
  #pragma unroll 4
  for (int q = lane; q < H / 4; q += 32) {
    float4 w = Wv[q], x = xv[q];
    s += w.x * x.x + w.y * x.y + w.z * x.z + w.w * x.w;
  }
  s = wave_reduce(s);
  if (lane == 0) {
    float b = sk ? (sk_bih[row] + sk_bhh[row]) : (op_bih[row] + op_bhh[row]);
    (sk ? xb_sk : xb_op)[row] = s + b;
  }
}

// --------------------------------------------- recurrent GEMV via fp32 WMMA
// gates[8192] = Whh @ h + xb.  One wave per 16-row tile (512 tiles).
// A (16x4 f32): A[m][k] = h[kb+k] replicated over m.
//   layout: lanes 0-15 -> {K0,K1} in VGPR0/1; lanes 16-31 -> {K2,K3}.
// B (4x16 f32): B[k][n] = Whh[rowbase+n][kb+k].
//   layout: lane n(+16) holds {K0,K1} ({K2,K3}) for column n in VGPR0/1.
// D[m][n] identical over m = partial dot of output row rowbase+n.
__global__ __launch_bounds__(256) void gemv_gates_wmma(
    const float* __restrict__ Whh, const float* __restrict__ xb,
    const float* __restrict__ h, float* __restrict__ gates) {
  __shared__ float hs[H];
  for (int t = threadIdx.x; t < H / 4; t += blockDim.x)
    ((float4*)hs)[t] = ((const float4*)h)[t];
  __syncthreads();

  int wave = threadIdx.x >> 5;
  int lane = threadIdx.x & 31;
  int tile = blockIdx.x * 8 + wave;          // 64 blocks * 8 waves = 512 tiles
  int rowbase = tile * 16;
  int n     = lane & 15;
  int khalf = (lane >> 4) << 1;              // 0 for lanes 0-15, 2 for 16-31
  const float* wrow = Whh + (size_t)(rowbase + n) * H + khalf;

  // 4 independent accumulators: WMMA i depends on WMMA i-4, not i-1.
  v8f acc0 = {}, acc1 = {}, acc2 = {}, acc3 = {};
  #pragma unroll 8
  for (int kb = 0; kb < H; kb += 4) {
    v2f a, b;
    a.x = hs[kb + khalf];
    a.y = hs[kb + khalf + 1];
    b.x = wrow[kb];
    b.y = wrow[kb + 1];
#if defined(__gfx1250__)
    if ((kb & 63) == 0)                      // prefetch weight stream ~1KB ahead
      __builtin_prefetch(wrow + kb + 256, 0, 3);
#endif
#if defined(__gfx1250__) && __has_builtin(__builtin_amdgcn_wmma_f32_16x16x4_f32)
    switch ((kb >> 2) & 3) {
      case 0:
        acc0 = __builtin_amdgcn_wmma_f32_16x16x4_f32(false, a, false, b,
                                                     (short)0, acc0, false, false);
        break;
      case 1:
        acc1 = __builtin_amdgcn_wmma_f32_16x16x4_f32(false, a, false, b,
                                                     (short)0, acc1, false, false);
        break;
      case 2:
        acc2 = __builtin_amdgcn_wmma_f32_16x16x4_f32(false, a, false, b,
                                                     (short)0, acc2, false, false);
        break;
      default:
        acc3 = __builtin_amdgcn_wmma_f32_16x16x4_f32(false, a, false, b,
                                                     (short)0, acc3, false, false);
        break;
    }
#else
    // Cross-lane fallback (also host pass): lane n accumulates K0..K3 terms
    // by folding in the partner half-wave's K2/K3 partial.
    {
      float p = a.x * b.x + a.y * b.y;
      acc0[0] += p + __shfl_xor(p, 16, 32);
    }
#endif
  }
  if (lane < 16) {
    int r = rowbase + lane;                   // D[M=0][N=lane] lives in acc[0]
    gates[r] = (acc0[0] + acc1[0]) + (acc2[0] + acc3[0]) + xb[r];
  }
}

// ----------------------------------------------------------- LSTM pointwise
// PyTorch gate order [i, f, g, o]
__global__ __launch_bounds__(256) void lstm_pointwise(
    const float* __restrict__ gates, float* __restrict__ h, float* __restrict__ c) {
  int j = blockIdx.x * blockDim.x + threadIdx.x;
  if (j < H) {
    float ig = gates[j];
    float fg = gates[j + H];
    float gg = gates[j + 2 * H];
    float og = gates[j + 3 * H];
    float si = 1.0f / (1.0f + expf(-ig));
    float sf = 1.0f / (1.0f + expf(-fg));
    float so = 1.0f / (1.0f + expf(-og));
    float c2 = sf * c[j] + si * tanhf(gg);
    c[j] = c2;
    h[j] = so * tanhf(c2);
  }
}

// --------------------------------------- 512-row GEMV (W_prev / W_curr @ h)
__global__ __launch_bounds__(256) void gemv512_kernel(
    const float* __restrict__ W, const float* __restrict__ h,
    float* __restrict__ out) {
  int gw   = (blockIdx.x * blockDim.x + threadIdx.x) >> 5;   // 0..511
  int lane = threadIdx.x & 31;
  const float4* Wv = (const float4*)(W + (size_t)gw * H);
  const float4* hv = (const float4*)h;
  float s = 0.0f;
  #pragma unroll 4
  for (int q = lane; q < H / 4; q += 32) {
    float4 w = Wv[q], x = hv[q];
    s += w.x * x.x + w.y * x.y + w.z * x.z + w.w * x.w;
  }
  s = wave_reduce(s);
  if (lane == 0) out[gw] = s;
}

// ------------------------------------------- softmax(out_W @ h + out_b) row
__global__ __launch_bounds__(192) void ops_row_kernel(
    const float* __restrict__ out_W, const float* __restrict__ out_b,
    const float* __restrict__ h, float* __restrict__ dst) {
  __shared__ float logits[NUM_OPS];
  int wave = threadIdx.x >> 5, lane = threadIdx.x & 31;
  if (wave < NUM_OPS) {
    const float4* Wv = (const float4*)(out_W + (size_t)wave * H);
    const float4* hv = (const float4*)h;
    float s = 0.0f;
    #pragma unroll 4
    for (int q = lane; q < H / 4; q += 32) {
      float4 w = Wv[q], x = hv[q];
      s += w.x * x.x + w.y * x.y + w.z * x.z + w.w * x.w;
    }
    s = wave_reduce(s);
    if (lane == 0) logits[wave] = s + out_b[wave];
  }
  __syncthreads();
  if (threadIdx.x == 0) {
    float m = logits[0];
    for (int j = 1; j < NUM_OPS; j++) m = fmaxf(m, logits[j]);
    float e[NUM_OPS], sum = 0.0f;
    for (int j = 0; j < NUM_OPS; j++) { e[j] = expf(logits[j] - m); sum += e[j]; }
    for (int j = 0; j < NUM_OPS; j++) dst[j] = e[j] / sum;
  }
}

// ---------------------- sigmoid( v . tanh(prev_proj[j] + curr) ), j = blockIdx
__global__ __launch_bounds__(256) void skip_scores_kernel(
    const float* __restrict__ prev, const float* __restrict__ curr,
    const float* __restrict__ v, float* __restrict__ dst) {
  __shared__ float red[8];
  int j = blockIdx.x;
  const float* pj = prev + (size_t)j * DIM_W;
  float s = 0.0f;
  for (int d = threadIdx.x; d < DIM_W; d += 256)
    s += v[d] * tanhf(pj[d] + curr[d]);
  int lane = threadIdx.x & 31, wave = threadIdx.x >> 5;
  s = wave_reduce(s);
  if (lane == 0) red[wave] = s;
  __syncthreads();
  if (threadIdx.x == 0) {
    float tot = 0.0f;
    for (int w = 0; w < 8; w++) tot += red[w];
    dst[j] = 1.0f / (1.0f + expf(-tot));
  }
}

// ---------------------------------------------------------------------------
extern "C" void kernel_launch(void* const* d_in, const int* in_sizes, int n_in,
                              void* d_out, int out_size, void* d_ws, size_t ws_size,
                              hipStream_t stream) {
  const float* g_emb  = (const float*)d_in[0];
  const float* op_Wih = (const float*)d_in[1];
  const float* op_Whh = (const float*)d_in[2];
  const float* op_bih = (const float*)d_in[3];
  const float* op_bhh = (const float*)d_in[4];
  const float* sk_Wih = (const float*)d_in[5];
  const float* sk_Whh = (const float*)d_in[6];
  const float* sk_bih = (const float*)d_in[7];
  const float* sk_bhh = (const float*)d_in[8];
  const float* out_W  = (const float*)d_in[9];
  const float* out_b  = (const float*)d_in[10];
  const float* W_prev = (const float*)d_in[11];
  const float* W_curr = (const float*)d_in[12];
  const float* v      = (const float*)d_in[13];
  float* out = (float*)d_out;

  // Workspace layout (floats): 41472 floats = ~162 KB.
  float* ws    = (float*)d_ws;
  float* h     = ws;            // 2048
  float* c     = ws + 2048;     // 2048
  float* gates = ws + 4096;     // 8192
  float* xb_op = ws + 12288;    // 8192
  float* xb_sk = ws + 20480;    // 8192
  float* prev  = ws + 28672;    // 24 * 512 (entry 0 stays zero)
  float* curr  = ws + 40960;    // 512
  const int WS_FLOATS = 41472;

  // Zero h, c, prev_projs (and everything else for determinism).
  zero_kernel<<<(WS_FLOATS + 255) / 256, 256, 0, stream>>>(ws, WS_FLOATS);

  // Constant input-projection + biases for both cell types (16384 waves).
  xb_kernel<<<2048, 256, 0, stream>>>(g_emb, op_Wih, op_bih, op_bhh,
                                      sk_Wih, sk_bih, sk_bhh, xb_op, xb_sk);

  for (int ci = 0; ci < 2 * NUM_NODES; ci++) {
    int i = ci / 2;
    bool odd = (ci & 1) != 0;

    // gates = Whh @ h + (x@Wih.T + bih + bhh)   [WMMA GEMV, 512 tiles]
    gemv_gates_wmma<<<64, 256, 0, stream>>>(odd ? sk_Whh : op_Whh,
                                            odd ? xb_sk : xb_op, h, gates);
    lstm_pointwise<<<(H + 255) / 256, 256, 0, stream>>>(gates, h, c);

    if (!odd) {
      ops_row_kernel<<<1, 192, 0, stream>>>(out_W, out_b, h, out + i * NUM_OPS);
    } else if (i > 0) {
      gemv512_kernel<<<64, 256, 0, stream>>>(W_curr, h, curr);
      skip_scores_kernel<<<i, 256, 0, stream>>>(prev, curr, v,
                                                out + NUM_NODES * NUM_OPS +
                                                    i * (i - 1) / 2);
    }
    // prev_projs[ci+1] = W_prev @ h  (only entries 1..22 are ever consumed)
    if (ci <= 21)
      gemv512_kernel<<<64, 256, 0, stream>>>(W_prev, h, prev + (size_t)(ci + 1) * DIM_W);
  }
}